// SpKBGATModified_59098749993532
// MI455X (gfx1250) — compile-verified
//
#include <hip/hip_runtime.h>
#include <math.h>

// ---------------- problem constants (from reference setup_inputs) ------------
constexpr int N_NODES = 50000;
constexpr int N_REL   = 500;
constexpr int D       = 200;   // feature dim (also NHID*H)
constexpr int NHID    = 100;
constexpr int H       = 2;
constexpr int NE      = 150000;
constexpr int NENH    = 30000;
constexpr int EA      = NE + NENH;   // 180000 edges total
constexpr int WAVE    = 32;          // gfx1250 is wave32-only

typedef __attribute__((ext_vector_type(2))) float v2f;
typedef __attribute__((ext_vector_type(4))) float v4f;
typedef __attribute__((ext_vector_type(8))) float v8f;

__device__ __forceinline__ float waveSum(float v) {
#pragma unroll
    for (int o = 16; o > 0; o >>= 1) v += __shfl_xor(v, o, WAVE);
    return v;
}

// ---------------- x = row / max(||row||, 1e-12) ------------------------------
__global__ void l2norm_rows(const float* __restrict__ src, float* __restrict__ dst,
                            int n, int d) {
    int wid  = (blockIdx.x * blockDim.x + threadIdx.x) >> 5;
    int lane = threadIdx.x & 31;
    if (wid >= n) return;
    const float* r = src + (size_t)wid * d;
    float ss = 0.f;
    for (int c = lane; c < d; c += WAVE) { float v = r[c]; ss += v * v; }
    ss = waveSum(ss);
    float inv = 1.0f / fmaxf(sqrtf(ss), 1e-12f);
    float* o = dst + (size_t)wid * d;
    for (int c = lane; c < d; c += WAVE) o[c] = r[c] * inv;
}

// ---------------- dst[c][r] = src[r][c] for 200x200 weights ------------------
__global__ void transpose200(const float* __restrict__ src, float* __restrict__ dst) {
    int i = blockIdx.x * blockDim.x + threadIdx.x;
    if (i >= D * D) return;
    int r = i / D, c = i % D;
    dst[(size_t)c * D + r] = src[i];
}

// ---------------- WMMA f32 strip GEMM (double-buffered K pipeline) -----------
// C[M][Nc] = X[M][ldx] * W[Nc][ldw]^T  (both row-major, K contiguous).
// One wave per 16-row strip of C; all NTT 16-col tiles accumulated in
// registers so each A fragment is loaded once per k-step. Out-of-range
// rows/cols are CLAMPED (not predicated): an A row only feeds C rows of the
// same index and a B column only feeds the same C column, so clamped garbage
// is never stored. K loop is software-pipelined: fragments for step k+4 are
// loaded into a second register set while step k's WMMAs execute, so the
// compiler can issue load clauses and avoid s_wait_loadcnt 0 before each WMMA.
template <int NTT>
__global__ void gemm_wmma_strip(const float* __restrict__ X, int ldx,
                                const float* __restrict__ W, int ldw, int woff,
                                float* __restrict__ C, int ldc,
                                int M, int Nc, int K) {
    int Mt = (M + 15) >> 4;
    int strip = blockIdx.x * (blockDim.x >> 5) + (threadIdx.x >> 5);
    if (strip >= Mt) return;
    int m0   = strip << 4;
    int lane = threadIdx.x & 31;
    int half = lane >> 4, lm = lane & 15;
    int koff = 2 * half;                                       // lane's K phase

    int arow = m0 + lm; if (arow >= M) arow = M - 1;           // clamp
    const float* xrow = X + (size_t)arow * ldx + koff;

    const float* wrow[NTT];
#pragma unroll
    for (int t = 0; t < NTT; ++t) {
        int c = t * 16 + lm; if (c >= Nc) c = Nc - 1;          // clamp
        wrow[t] = W + (size_t)c * ldw + woff + koff;
    }

    v8f acc[NTT];
#pragma unroll
    for (int t = 0; t < NTT; ++t) acc[t] = (v8f){0.f,0.f,0.f,0.f,0.f,0.f,0.f,0.f};

    // prologue: load fragments for k = 0
    v2f a0 = *(const v2f*)(xrow);
    v2f b0[NTT];
#pragma unroll
    for (int t = 0; t < NTT; ++t) b0[t] = *(const v2f*)(wrow[t]);

    for (int k = 0; k < K; k += 4) {
        int kn = k + 4;
        v2f a1 = a0;
        v2f b1[NTT];
#pragma unroll
        for (int t = 0; t < NTT; ++t) b1[t] = b0[t];
        if (kn < K) {                                          // prefetch k+4
            a1 = *(const v2f*)(xrow + kn);
#pragma unroll
            for (int t = 0; t < NTT; ++t) b1[t] = *(const v2f*)(wrow[t] + kn);
        }
#pragma unroll
        for (int t = 0; t < NTT; ++t)
            acc[t] = __builtin_amdgcn_wmma_f32_16x16x4_f32(false, a0, false, b0[t],
                                                           (short)0, acc[t],
                                                           false, false);
        a0 = a1;
#pragma unroll
        for (int t = 0; t < NTT; ++t) b0[t] = b1[t];
    }

#pragma unroll
    for (int t = 0; t < NTT; ++t) {
        int col = t * 16 + lm;
        if (col < Nc) {
#pragma unroll
            for (int r = 0; r < 8; ++r) {
                int row = m0 + r + (half ? 8 : 0);
                if (row < M) C[(size_t)row * ldc + col] = acc[t][r];
            }
        }
    }
}

static inline void launch_gemm_nt(const float* X, int ldx,
                                  const float* W, int ldw, int woff,
                                  float* C, int ldc, int M, int Nc, int K,
                                  hipStream_t s) {
    int Mt = (M + 15) / 16;
    int blocks = (Mt + 7) / 8;                    // 8 waves / 256-thread block
    int Nt = (Nc + 15) / 16;
    if (Nt <= 7) gemm_wmma_strip<7 ><<<blocks, 256, 0, s>>>(X, ldx, W, ldw, woff, C, ldc, M, Nc, K);
    else         gemm_wmma_strip<13><<<blocks, 256, 0, s>>>(X, ldx, W, ldw, woff, C, ldc, M, Nc, K);
}

// ---------------- fused per-edge GAT pass -----------------------------------
// One wave per edge. m = Ap[dst] + Bp[src] + relp[r1] (+ relp[r2] for n-hop);
// e = exp(-leakyrelu(m . a2, 0.2)); atomically accumulate e and e*m into dst.
// float4 loads: rows are 400B/800B and 16B-aligned.
__global__ void edge_gat(const float* __restrict__ Ap, const float* __restrict__ Bp,
                         const float* __restrict__ relp, const float* __restrict__ a2,
                         const int* __restrict__ elist, const int* __restrict__ etype,
                         const int* __restrict__ nhop,
                         float* __restrict__ rowsum, float* __restrict__ hp, int F) {
    int eid  = (blockIdx.x * blockDim.x + threadIdx.x) >> 5;
    if (eid >= EA) return;
    int lane = threadIdx.x & 31;

    int dst, src, r1, r2;
    if (eid < NE) {
        dst = elist[eid]; src = elist[NE + eid]; r1 = etype[eid]; r2 = -1;
    } else {
        int j = eid - NE;
        dst = nhop[4 * j + 3]; src = nhop[4 * j + 0];
        r1  = nhop[4 * j + 1]; r2  = nhop[4 * j + 2];
    }

    const float* ad = Ap   + (size_t)dst * F;
    const float* bs = Bp   + (size_t)src * F;
    const float* rA = relp + (size_t)r1  * F;
    const float* rB = relp + (size_t)((r2 >= 0) ? r2 : r1) * F;

    v4f m[2] = {(v4f){0.f,0.f,0.f,0.f}, (v4f){0.f,0.f,0.f,0.f}};
    float dp = 0.f;
#pragma unroll
    for (int i = 0; i < 2; ++i) {
        int c = 4 * (lane + WAVE * i);
        if (c < F) {
            v4f mv = *(const v4f*)(ad + c) + *(const v4f*)(bs + c)
                   + *(const v4f*)(rA + c);
            if (r2 >= 0) mv += *(const v4f*)(rB + c);
            v4f a4 = *(const v4f*)(a2 + c);
            dp += mv.x * a4.x + mv.y * a4.y + mv.z * a4.z + mv.w * a4.w;
            m[i] = mv;
        }
    }
    dp = waveSum(dp);
    float p  = dp > 0.f ? dp : 0.2f * dp;            // leaky_relu, alpha=0.2
    float ee = __expf(-p);
    if (lane == 0) atomicAdd(&rowsum[dst], ee);

    float* hd = hp + (size_t)dst * F;
#pragma unroll
    for (int i = 0; i < 2; ++i) {
        int c = 4 * (lane + WAVE * i);
        if (c < F) {
            atomicAdd(hd + c + 0, ee * m[i].x);
            atomicAdd(hd + c + 1, ee * m[i].y);
            atomicAdd(hd + c + 2, ee * m[i].z);
            atomicAdd(hd + c + 3, ee * m[i].w);
        }
    }
}

// ---------------- x1[n][h*100+k] = elu(hp1[h][n][k] / rowsum1[h][n]) ---------
__global__ void build_x1(const float* __restrict__ hp1, const float* __restrict__ rs1,
                         float* __restrict__ x1) {
    int i = blockIdx.x * blockDim.x + threadIdx.x;
    if (i >= N_NODES * (H * NHID)) return;
    int n = i / (H * NHID), c = i % (H * NHID);
    int h = c / NHID, k = c % NHID;
    float rs = rs1[h * N_NODES + n];
    rs = (rs == 0.f) ? 1e-12f : rs;
    float v = hp1[((size_t)h * N_NODES + n) * NHID + k] / rs;
    x1[i] = v > 0.f ? v : (__expf(v) - 1.f);
}

__global__ void scatter_mask(const int* __restrict__ idx, float* __restrict__ mask, int n) {
    int i = blockIdx.x * blockDim.x + threadIdx.x;
    if (i < n) mask[idx[i]] = 1.0f;
}

// ---------------- out_ent = l2norm(xW + mask * elu(hp2/rowsum2)) -------------
__global__ void finalize(const float* __restrict__ xW, const float* __restrict__ hp2,
                         const float* __restrict__ rs2, const float* __restrict__ mask,
                         float* __restrict__ out) {
    int wid  = (blockIdx.x * blockDim.x + threadIdx.x) >> 5;
    if (wid >= N_NODES) return;
    int lane = threadIdx.x & 31;
    float rs = rs2[wid]; rs = (rs == 0.f) ? 1e-12f : rs;
    float mk = mask[wid];
    v4f t[2] = {(v4f){0.f,0.f,0.f,0.f}, (v4f){0.f,0.f,0.f,0.f}};
    float ss = 0.f;
#pragma unroll
    for (int i = 0; i < 2; ++i) {
        int c = 4 * (lane + WAVE * i);
        if (c < D) {
            v4f hv = *(const v4f*)(hp2 + (size_t)wid * D + c);
            v4f xv = *(const v4f*)(xW  + (size_t)wid * D + c);
            v4f v;
            float h0 = hv.x / rs, h1 = hv.y / rs, h2 = hv.z / rs, h3 = hv.w / rs;
            v.x = xv.x + mk * (h0 > 0.f ? h0 : (__expf(h0) - 1.f));
            v.y = xv.y + mk * (h1 > 0.f ? h1 : (__expf(h1) - 1.f));
            v.z = xv.z + mk * (h2 > 0.f ? h2 : (__expf(h2) - 1.f));
            v.w = xv.w + mk * (h3 > 0.f ? h3 : (__expf(h3) - 1.f));
            ss += v.x * v.x + v.y * v.y + v.z * v.z + v.w * v.w;
            t[i] = v;
        }
    }
    ss = waveSum(ss);
    float inv = 1.0f / fmaxf(sqrtf(ss), 1e-12f);
#pragma unroll
    for (int i = 0; i < 2; ++i) {
        int c = 4 * (lane + WAVE * i);
        if (c < D) {
            v4f v = t[i];
            v.x *= inv; v.y *= inv; v.z *= inv; v.w *= inv;
            *(v4f*)(out + (size_t)wid * D + c) = v;
        }
    }
}

// ---------------- orchestration ---------------------------------------------
extern "C" void kernel_launch(void* const* d_in, const int* in_sizes, int n_in,
                              void* d_out, int out_size, void* d_ws, size_t ws_size,
                              hipStream_t stream) {
    (void)n_in; (void)out_size;
    const float* ent  = (const float*)d_in[0];   // [N,200]
    const float* rel  = (const float*)d_in[1];   // [500,200]
    const float* Went = (const float*)d_in[2];   // [200,200]
    const float* Wgat = (const float*)d_in[3];   // [200,200]
    const float* aH   = (const float*)d_in[4];   // [2][100][600]
    const float* a2H  = (const float*)d_in[5];   // [2][1][100]
    const float* aO   = (const float*)d_in[6];   // [200][600]
    const float* a2O  = (const float*)d_in[7];   // [1][200]
    const int*   batch = (const int*)d_in[8];    // [20000]
    const int*   elist = (const int*)d_in[9];    // [2,150000]
    const int*   etype = (const int*)d_in[10];   // [150000]
    const int*   nhop  = (const int*)d_in[11];   // [30000,4]
    int nbatch = in_sizes[8];

    float* out_ent = (float*)d_out;                         // [N,200]
    float* out_rel = out_ent + (size_t)N_NODES * D;         // [500,200]
    float* mask    = out_rel + (size_t)N_REL * D;           // [N]

    // workspace (float offsets); total 50,430,000 floats ~= 201.7 MB
    float* w   = (float*)d_ws;
    float* X   = w;              // 10M : l2norm(x)
    float* BA  = w + 10000000;   // 10M : A1[h][n][100]  -> later P1[n][200]
    float* BB  = w + 20000000;   // 10M : B1             -> later P2
    float* HP  = w + 30000000;   // 10M : hp1[h][n][100] -> later hp2[n][200]
    float* XC  = w + 40000000;   // 10M : x1[n][200]     -> later xW[n][200]
    float* RP1 = w + 50000000;   // 100k: relproj1[h][500][100]
    float* RP2 = w + 50100000;   // 100k: relproj2[500][200]
    float* RS1 = w + 50200000;   // 100k: rowsum1[h][n]
    float* RS2 = w + 50300000;   //  50k: rowsum2[n]
    float* WGT = w + 50350000;   //  40k: W_gat^T
    float* WET = w + 50390000;   //  40k: W_entities^T
    if (ws_size < (size_t)50430000 * sizeof(float)) return;

    // 0) transpose the two square weights so all GEMMs use the NT fast path
    transpose200<<<(D * D + 255) / 256, 256, 0, stream>>>(Wgat, WGT);
    transpose200<<<(D * D + 255) / 256, 256, 0, stream>>>(Went, WET);

    // 1) x = l2norm(entity_embeddings)
    l2norm_rows<<<(N_NODES + 7) / 8, 256, 0, stream>>>(ent, X, N_NODES, D);

    // 2) layer-1 projections (WMMA): A1/B1 per head, relation projections
    for (int h = 0; h < H; ++h) {
        const float* ah = aH + (size_t)h * NHID * 600;      // [100][600]
        launch_gemm_nt(X,   D, ah, 600, 0,   BA + (size_t)h * N_NODES * NHID, NHID,
                       N_NODES, NHID, D, stream);
        launch_gemm_nt(X,   D, ah, 600, 200, BB + (size_t)h * N_NODES * NHID, NHID,
                       N_NODES, NHID, D, stream);
        launch_gemm_nt(rel, D, ah, 600, 400, RP1 + (size_t)h * N_REL * NHID, NHID,
                       N_REL, NHID, D, stream);
    }

    // 3) edge pass, layer 1 (per head)
    hipMemsetAsync(HP,  0, (size_t)H * N_NODES * NHID * sizeof(float), stream);
    hipMemsetAsync(RS1, 0, (size_t)H * N_NODES * sizeof(float), stream);
    for (int h = 0; h < H; ++h) {
        edge_gat<<<(EA + 7) / 8, 256, 0, stream>>>(
            BA + (size_t)h * N_NODES * NHID, BB + (size_t)h * N_NODES * NHID,
            RP1 + (size_t)h * N_REL * NHID, a2H + (size_t)h * NHID,
            elist, etype, nhop,
            RS1 + (size_t)h * N_NODES, HP + (size_t)h * N_NODES * NHID, NHID);
    }

    // 4) x1 = elu(hp1/rowsum1), head-interleaved [n][h*100+k]
    build_x1<<<(N_NODES * H * NHID + 255) / 256, 256, 0, stream>>>(HP, RS1, XC);

    // 5) out_rel = rel @ W_gat = rel * WGT^T (straight into d_out)
    launch_gemm_nt(rel, D, WGT, D, 0, out_rel, D, N_REL, D, D, stream);

    // 6) layer-2 projections: relp2 = out_rel @ aO[:,400:600]^T ; P1/P2 from x1
    launch_gemm_nt(out_rel, D, aO, 600, 400, RP2, D, N_REL, D, D, stream);
    launch_gemm_nt(XC, D, aO, 600, 0,   BA, D, N_NODES, D, D, stream);
    launch_gemm_nt(XC, D, aO, 600, 200, BB, D, N_NODES, D, D, stream);

    // 7) edge pass, layer 2 (F = 200)
    hipMemsetAsync(HP,  0, (size_t)N_NODES * D * sizeof(float), stream);
    hipMemsetAsync(RS2, 0, (size_t)N_NODES * sizeof(float), stream);
    edge_gat<<<(EA + 7) / 8, 256, 0, stream>>>(BA, BB, RP2, a2O,
                                               elist, etype, nhop, RS2, HP, D);

    // 8) xW = x @ W_entities = x * WET^T (reuse XC — x1 is dead now)
    launch_gemm_nt(X, D, WET, D, 0, XC, D, N_NODES, D, D, stream);

    // 9) mask
    hipMemsetAsync(mask, 0, (size_t)N_NODES * sizeof(float), stream);
    scatter_mask<<<(nbatch + 255) / 256, 256, 0, stream>>>(batch, mask, nbatch);

    // 10) out_ent = l2norm(xW + mask * elu(hp2/rowsum2))
    finalize<<<(N_NODES + 7) / 8, 256, 0, stream>>>(XC, HP, RS2, mask, out_ent);
}